// EdgeNetworkMessage_23304492548529
// MI455X (gfx1250) — compile-verified
//
#include <hip/hip_runtime.h>

// ---------------------------------------------------------------------------
// EdgeNetworkMessage fused kernel for gfx1250 (MI455X, wave32, WMMA).
//
//   hdn = relu(ef @ W1 + b1)            (E,16)->(E,64)
//   A   = hdn @ W2 + b2                 (E,64)->(E,1024)   [never materialized]
//   msg[e] = A[e].reshape(32,32) @ h_w[e]                  (E,32)
//
// Roofline: 33.8 GFLOP f32, mandatory HBM traffic ~80MB (3.4us @ 23.3TB/s).
// Fully fused -> compute-bound on the f32 matrix pipe (V_WMMA_F32_16X16X4_F32,
// reference-precision). Stage-B issue overhead minimized:
//   - W2 pre-transposed once into d_ws so each B-operand K-pair is one b64 load
//   - b1/b2 biases seeded into the WMMA accumulator chain (zero VALU cost)
//   - stage-C slab spilled as b64 pairs in a column-interleaved LDS layout
// ---------------------------------------------------------------------------

typedef __attribute__((ext_vector_type(2))) float v2f;
typedef __attribute__((ext_vector_type(4))) float v4f;
typedef __attribute__((ext_vector_type(8))) float v8f;

namespace {
constexpr int kE   = 250000;
constexpr int kD   = 32;            // HIDDEN_DIM
constexpr int kED  = 16;            // EDGE_DIM
constexpr int kEH  = 64;            // EDGE_HIDDEN
constexpr int kDD  = kD * kD;       // 1024
constexpr int kTiles = kE / 16;     // 15625 (exact)
constexpr int kMT    = 2;           // 16-edge M-tiles per wave
constexpr int kWaves = 8;           // waves per block (256 threads)
constexpr int kTilesPerBlock = kMT * kWaves;                          // 16
constexpr int kGrid = (kTiles + kTilesPerBlock - 1) / kTilesPerBlock; // 977
constexpr size_t kW2TBytes = (size_t)kEH * kDD * sizeof(float);       // 256KB
} // namespace

static __device__ __forceinline__ v8f wmma_k4(v2f a, v2f b, v8f c) {
  // 8-arg pattern: (neg_a, A, neg_b, B, c_mod, C, reuse_a, reuse_b)
  return __builtin_amdgcn_wmma_f32_16x16x4_f32(false, a, false, b,
                                               (short)0, c, false, false);
}

static __device__ __forceinline__ v8f splat8(float x) {
  return (v8f){x, x, x, x, x, x, x, x};
}

// W2T[(kk*1024 + col)*4 + q] = W2[(kk*4 + q)*1024 + col]
// -> a lane's B-operand K-pair (rows kk*4+2*half, +1) becomes one b64.
__global__ __launch_bounds__(256)
void transpose_w2(const float* __restrict__ W2, float* __restrict__ W2T) {
  const int idx = blockIdx.x * 256 + threadIdx.x;       // 65536 elements
  const int q   = idx & 3;
  const int col = (idx >> 2) & (kDD - 1);
  const int kk  = idx >> 12;
  W2T[idx] = W2[(kk * 4 + q) * kDD + col];
}

template <bool USE_W2T>
__global__ __launch_bounds__(256)
void edge_msg_fused(const float* __restrict__ h_w,
                    const float* __restrict__ ef,
                    const float* __restrict__ W1,
                    const float* __restrict__ b1,
                    const float* __restrict__ W2,   // transposed iff USE_W2T
                    const float* __restrict__ b2,
                    float* __restrict__ out)
{
  // 4KB scratch per wave (stage-A layout shuffle / stage-C slab). Private per
  // wave -> no barriers; same-wave LDS ordering via s_wait_dscnt.
  __shared__ float lds[kWaves * 1024];

  const int lane = threadIdx.x & 31;
  const int wv   = threadIdx.x >> 5;
  float* scratch = &lds[wv * 1024];

  const int half = lane >> 4;   // lane half: carries K pairs (A/B), M+8 (C/D)
  const int l16  = lane & 15;

  // --- tile assignment: wave gw owns tiles 2*gw, 2*gw+1 (clamped tail) ----
  const int gw = blockIdx.x * kWaves + wv;
  int  tile[kMT];
  bool tvalid[kMT];
#pragma unroll
  for (int m = 0; m < kMT; ++m) {
    const int t = gw * kMT + m;
    tvalid[m] = (t < kTiles);
    tile[m]   = tvalid[m] ? t : (kTiles - 1);   // clamp: uniform control flow
  }

  // --- per-lane h_w row, split into lo/hi column halves (j<16 / j>=16) to
  //     match the interleaved stage-C slab layout; resident in registers. ---
  const int eRow = tile[half] * 16 + l16;       // lane 'lane' owns this edge
  float hwlo[16], hwhi[16];
#pragma unroll
  for (int t2 = 0; t2 < 4; ++t2) {
    *(v4f*)&hwlo[4 * t2] = *(const v4f*)(h_w + eRow * kD + 4 * t2);
    *(v4f*)&hwhi[4 * t2] = *(const v4f*)(h_w + eRow * kD + 16 + 4 * t2);
  }

  // =========================================================================
  // Stage A: hdn = relu(ef @ W1 + b1), per M-tile; keep the 16 stage-B
  // A-operand v2f's per tile resident in VGPRs.
  // A 16x4 f32 layout (ISA 7.12.2): lane L -> row M=L%16,
  //   VGPR0 = K=kbase+2*(L/16), VGPR1 = K+1.  B 4x16 mirrors (N=l16).
  // C/D 16x16: VGPR r -> (M = r + 8*half, N = l16).
  // =========================================================================
  v2f Aop[kMT][16];

#pragma unroll
  for (int m = 0; m < kMT; ++m) {
    const int eb = tile[m] * 16;

    v2f aef[4];
#pragma unroll
    for (int k4 = 0; k4 < 4; ++k4)
      aef[k4] = *(const v2f*)(ef + (eb + l16) * kED + k4 * 4 + 2 * half);

#pragma unroll
    for (int nt = 0; nt < 4; ++nt) {            // 4 N-tiles cover EH=64
      v8f c = splat8(b1[nt * 16 + l16]);        // bias seeded into the chain
#pragma unroll
      for (int k4 = 0; k4 < 4; ++k4) {          // K=16 in 4 chained WMMAs
        const int krow = k4 * 4 + 2 * half;
        v2f bv = { W1[krow * kEH + nt * 16 + l16],
                   W1[(krow + 1) * kEH + nt * 16 + l16] };
        c = wmma_k4(aef[k4], bv, c);
      }
#pragma unroll
      for (int r = 0; r < 8; ++r) {             // ReLU, C/D layout -> LDS
        const float v = c[r] > 0.f ? c[r] : 0.f;
        scratch[(r + 8 * half) * kEH + nt * 16 + l16] = v;
      }
    }

    // Re-read hdn in A-operand layout (contiguous K pair -> one b64 load).
#pragma unroll
    for (int kk = 0; kk < 16; ++kk)
      Aop[m][kk] = *(const v2f*)&scratch[l16 * kEH + kk * 4 + 2 * half];
  }

  // =========================================================================
  // Stage B + C fused over the 32 rows i of the per-edge 32x32 matrix:
  // 64 chained WMMAs produce the 32-edge x 32-col slab of A (bias pre-seeded),
  // the slab is spilled to LDS as b64 column-pairs, then each lane contracts
  // its edge's row against h_w. W2T is L2/WGP$-resident (256KB).
  // =========================================================================
  float msgbuf[4];

#pragma unroll 1
  for (int i = 0; i < kD; ++i) {
    const float bb0 = b2[i * kD + l16];          // col j = l16
    const float bb1 = b2[i * kD + 16 + l16];     // col j = 16 + l16
    v8f a00 = splat8(bb0), a01 = splat8(bb1);    // Mtile0: cols [0,16), [16,32)
    v8f a10 = splat8(bb0), a11 = splat8(bb1);    // Mtile1

#pragma unroll
    for (int kk = 0; kk < 16; ++kk) {            // K = 64
      v2f bv0, bv1;
      if constexpr (USE_W2T) {
        const float* p = W2 + (kk * kDD + i * kD + l16) * 4 + 2 * half;
        bv0 = *(const v2f*)p;                    // K-pair, col l16
        bv1 = *(const v2f*)(p + 64);             // K-pair, col 16+l16
      } else {
        const float* p = W2 + (kk * 4 + 2 * half) * kDD + i * kD + l16;
        bv0 = (v2f){ p[0],  p[kDD] };
        bv1 = (v2f){ p[16], p[kDD + 16] };
      }
      // 4 independent accumulators interleaved -> hides WMMA RAW latency.
      a00 = wmma_k4(Aop[0][kk], bv0, a00);
      a10 = wmma_k4(Aop[1][kk], bv0, a10);
      a01 = wmma_k4(Aop[0][kk], bv1, a01);
      a11 = wmma_k4(Aop[1][kk], bv1, a11);
    }

    // Spill slab interleaved: slab[row][2*c + p] = A(row, j = p*16 + c),
    // so each lane's (a_m0[r], a_m1[r]) pair is one ds_store_b64.
#pragma unroll
    for (int r = 0; r < 8; ++r) {
      const int row = r + 8 * half;
      *(v2f*)&scratch[row * kD + 2 * l16]        = (v2f){a00[r], a01[r]};
      *(v2f*)&scratch[(16 + row) * kD + 2 * l16] = (v2f){a10[r], a11[r]};
    }

    // Stage C: lane 'lane' owns local edge 'lane' (row = lane). The v4f read
    // covers j = {2t, 16+2t, 2t+1, 17+2t}; dot order is commutative.
    float s = 0.f;
#pragma unroll
    for (int t = 0; t < 8; ++t) {
      v4f a4 = *(const v4f*)&scratch[lane * kD + 4 * t];
      s += a4.x * hwlo[2 * t]     + a4.y * hwhi[2 * t] +
           a4.z * hwlo[2 * t + 1] + a4.w * hwhi[2 * t + 1];
    }

    msgbuf[i & 3] = s;
    if ((i & 3) == 3 && tvalid[half]) {          // b128 store every 4 rows
      v4f o = { msgbuf[0], msgbuf[1], msgbuf[2], msgbuf[3] };
      *(v4f*)(out + eRow * kD + (i & ~3)) = o;
    }
  }
}

extern "C" void kernel_launch(void* const* d_in, const int* in_sizes, int n_in,
                              void* d_out, int out_size, void* d_ws, size_t ws_size,
                              hipStream_t stream) {
  // setup_inputs order: h_v(0, unused), h_w(1), edge_features(2),
  //                     W1(3), b1(4), W2(5), b2(6)
  const float* h_w = (const float*)d_in[1];
  const float* ef  = (const float*)d_in[2];
  const float* W1  = (const float*)d_in[3];
  const float* b1  = (const float*)d_in[4];
  const float* W2  = (const float*)d_in[5];
  const float* b2  = (const float*)d_in[6];
  float* out = (float*)d_out;

  dim3 grid(kGrid), block(kWaves * 32);

  if (ws_size >= kW2TBytes) {
    float* W2T = (float*)d_ws;
    hipLaunchKernelGGL(transpose_w2, dim3(kEH * kDD / 4 / 256), dim3(256),
                       0, stream, W2, W2T);
    hipLaunchKernelGGL((edge_msg_fused<true>), grid, block, 0, stream,
                       h_w, ef, W1, b1, W2T, b2, out);
  } else {
    hipLaunchKernelGGL((edge_msg_fused<false>), grid, block, 0, stream,
                       h_w, ef, W1, b1, W2, b2, out);
  }
}